// AttentionExample_21706764714808
// MI455X (gfx1250) — compile-verified
//
#include <hip/hip_runtime.h>
#include <math.h>

typedef __attribute__((ext_vector_type(2))) float v2f;
typedef __attribute__((ext_vector_type(8))) float v8f;

#define SEQ_LEN 32768
#define HIDDEN  1024
#define HCHUNK  64            // h-rows per partial block
#define NHG     (HIDDEN / HCHUNK)   // 16 h-groups

// ---------------------------------------------------------------------------
// Kernel 1a: partial fold of W and `other`:
//   partial[hg][k] = sum_{h in hg} other[h] * W[h, 1024+k]
// 64 blocks (4 k-groups x 16 h-groups) -> enough parallelism to stream 4 MB.
// (hidden, b, and W[:, :1024] only shift all scores by a constant, which
//  softmax cancels exactly — skipped.)
// ---------------------------------------------------------------------------
__global__ void __launch_bounds__(256)
attn_make_v_part(const float* __restrict__ W,
                 const float* __restrict__ other,
                 float* __restrict__ partial) {
  const int kg = blockIdx.x & 3;        // 4 k-groups of 256 columns
  const int hg = blockIdx.x >> 2;       // 16 h-groups of 64 rows
  const int k  = kg * 256 + threadIdx.x;
  const int h0 = hg * HCHUNK;
  float acc = 0.0f;
  #pragma unroll 8
  for (int h = h0; h < h0 + HCHUNK; ++h) {
    acc = fmaf(other[h], W[(size_t)h * (2 * HIDDEN) + HIDDEN + k], acc);
  }
  partial[hg * HIDDEN + k] = acc;
}

// Kernel 1b: v2[k] = sum_hg partial[hg][k]   (fixed order -> deterministic)
__global__ void __launch_bounds__(256)
attn_make_v_reduce(const float* __restrict__ partial,
                   float* __restrict__ v2) {
  const int k = blockIdx.x * 256 + threadIdx.x;
  float acc = 0.0f;
  #pragma unroll
  for (int hg = 0; hg < NHG; ++hg) acc += partial[hg * HIDDEN + k];
  v2[k] = acc;
}

// ---------------------------------------------------------------------------
// Kernel 2: scores[s] = enc[s,:] . v2   via V_WMMA_F32_16X16X4_F32.
// A fragment = 8B global load per lane (16 sequential row-streams per wave),
// B fragment = v2 broadcast from LDS (same index pattern, ds_load_b64).
// 8 waves/block = 128 rows/block; 256 blocks stream the 128 MB of enc.
// ---------------------------------------------------------------------------
__global__ void __launch_bounds__(256)
attn_gemv_wmma(const float* __restrict__ enc,
               const float* __restrict__ v2,
               float* __restrict__ scores) {
  __shared__ float vlds[HIDDEN];
  for (int i = threadIdx.x; i < HIDDEN; i += 256) vlds[i] = v2[i];
  __syncthreads();

  const int wave  = threadIdx.x >> 5;    // 0..7
  const int lane  = threadIdx.x & 31;    // wave32
  const int half  = lane >> 4;           // 0: K=0,1   1: K=2,3
  const int m     = lane & 15;           // row within 16-row tile
  const int rbase = blockIdx.x * 128 + wave * 16;

  const float* aptr = enc + (size_t)(rbase + m) * HIDDEN + 2 * half;
  const float* bptr = vlds + 2 * half;

  v8f c = {};
  #pragma unroll 8
  for (int k = 0; k < HIDDEN; k += 4) {
    v2f a = *(const v2f*)(aptr + k);     // global_load_b64
    v2f b = *(const v2f*)(bptr + k);     // ds_load_b64 (LDS broadcast)
    if ((k & 31) == 0 && k + 32 < HIDDEN)
      __builtin_prefetch(aptr + k + 32, 0, 3);   // global_prefetch_b8, next line
    // (neg_a, A, neg_b, B, c_mod, C, reuse_a, reuse_b)
    c = __builtin_amdgcn_wmma_f32_16x16x4_f32(
        false, a, false, b, (short)0, c, false, false);
  }

  // D layout: VGPR j -> M=j (lanes 0-15) / M=j+8 (lanes 16-31); all N equal.
  if (lane == 0) {
    #pragma unroll
    for (int j = 0; j < 8; ++j) scores[rbase + j] = c[j];
  } else if (lane == 16) {
    #pragma unroll
    for (int j = 0; j < 8; ++j) scores[rbase + 8 + j] = c[j];
  }
}

// ---------------------------------------------------------------------------
// Softmax, 3 parallel stages (all reduction orders fixed -> deterministic).
// ---------------------------------------------------------------------------
// 3a: per-block (256 elems) max + exp-sum
__global__ void __launch_bounds__(256)
attn_softmax_part(const float* __restrict__ scores,
                  float* __restrict__ pmax, float* __restrict__ psum) {
  __shared__ float red[256];
  const int t = threadIdx.x;
  const float x = scores[blockIdx.x * 256 + t];
  red[t] = x; __syncthreads();
  for (int s = 128; s > 0; s >>= 1) {
    if (t < s) red[t] = fmaxf(red[t], red[t + s]);
    __syncthreads();
  }
  const float m = red[0]; __syncthreads();
  red[t] = __expf(x - m); __syncthreads();
  for (int s = 128; s > 0; s >>= 1) {
    if (t < s) red[t] += red[t + s];
    __syncthreads();
  }
  if (t == 0) { pmax[blockIdx.x] = m; psum[blockIdx.x] = red[0]; }
}

// 3b: combine 128 partials -> global max M and 1/S
__global__ void __launch_bounds__(128)
attn_softmax_comb(const float* __restrict__ pmax,
                  const float* __restrict__ psum,
                  float* __restrict__ stats) {
  __shared__ float rm[128];
  __shared__ float rs[128];
  const int t = threadIdx.x;
  const float m = pmax[t];
  rm[t] = m; __syncthreads();
  for (int s = 64; s > 0; s >>= 1) {
    if (t < s) rm[t] = fmaxf(rm[t], rm[t + s]);
    __syncthreads();
  }
  const float M = rm[0]; __syncthreads();
  rs[t] = psum[t] * __expf(m - M); __syncthreads();
  for (int s = 64; s > 0; s >>= 1) {
    if (t < s) rs[t] += rs[t + s];
    __syncthreads();
  }
  if (t == 0) { stats[0] = M; stats[1] = 1.0f / rs[0]; }
}

// 3c: normalize + write output
__global__ void __launch_bounds__(256)
attn_softmax_write(const float* __restrict__ scores,
                   const float* __restrict__ stats,
                   float* __restrict__ out) {
  const int i = blockIdx.x * 256 + threadIdx.x;
  out[i] = __expf(scores[i] - stats[0]) * stats[1];
}

// ---------------------------------------------------------------------------
extern "C" void kernel_launch(void* const* d_in, const int* in_sizes, int n_in,
                              void* d_out, int out_size, void* d_ws, size_t ws_size,
                              hipStream_t stream) {
  // setup_inputs order: hidden, encoder_outputs, W, b, other
  const float* enc   = (const float*)d_in[1];  // (32768, 1, 1024)
  const float* W     = (const float*)d_in[2];  // (1024, 2048)
  const float* other = (const float*)d_in[4];  // (1, 1024)
  (void)in_sizes; (void)n_in; (void)out_size; (void)ws_size;

  float* out = (float*)d_out;                  // (1,1,32768) f32

  // workspace layout (floats): ~202 KB total
  float* v2      = (float*)d_ws;               // 1024
  float* scores  = v2 + HIDDEN;                // 32768
  float* partial = scores + SEQ_LEN;           // NHG*1024 = 16384
  float* pmax    = partial + NHG * HIDDEN;     // 128
  float* psum    = pmax + 128;                 // 128
  float* stats   = psum + 128;                 // 2

  attn_make_v_part  <<<4 * NHG,       256, 0, stream>>>(W, other, partial);
  attn_make_v_reduce<<<HIDDEN / 256,  256, 0, stream>>>(partial, v2);
  attn_gemv_wmma    <<<SEQ_LEN / 128, 256, 0, stream>>>(enc, v2, scores);
  attn_softmax_part <<<SEQ_LEN / 256, 256, 0, stream>>>(scores, pmax, psum);
  attn_softmax_comb <<<1,             128, 0, stream>>>(pmax, psum, stats);
  attn_softmax_write<<<SEQ_LEN / 256, 256, 0, stream>>>(scores, stats, out);
}